// CapsuleLayer_63050119905748
// MI455X (gfx1250) — compile-verified
//
#include <hip/hip_runtime.h>
#include <hip/hip_bf16.h>

typedef __attribute__((ext_vector_type(2))) float v2f;
typedef __attribute__((ext_vector_type(8))) float v8f;
typedef __attribute__((ext_vector_type(4))) int   v4i;

#define B_SZ   64
#define I_SZ   2048
#define P_SZ   16
#define J_SZ   32
#define D_SZ   32
#define NIBLK  16                   // i-blocks (grid.y)
#define IB_PER_WG (I_SZ / NIBLK)    // 128 capsules per workgroup
#define CHUNK  4                    // capsules staged per phase
#define ASTR   20                   // A row stride (floats): 16B-aligned + conflict-free
#define BSTR   36                   // B row stride (floats): 16B-aligned + conflict-free

#if __has_builtin(__builtin_amdgcn_global_load_async_to_lds_b128)
#define HAS_ASYNC 1
#else
#define HAS_ASYNC 0
#endif

#if HAS_ASYNC
__device__ __forceinline__ void g2l_async_b128(const float* g, float* l) {
    __builtin_amdgcn_global_load_async_to_lds_b128(
        (__attribute__((address_space(1))) v4i*)(g),
        (__attribute__((address_space(3))) v4i*)(l), 0, 0);
}
#endif

// ---------------------------------------------------------------------------
// Kernel 1: c[i,j] = softmax over j of bias[i,j].  One wave (32 lanes) per i.
// ---------------------------------------------------------------------------
__global__ void caps_softmax_kernel(const float* __restrict__ bias,
                                    float* __restrict__ c) {
    int gid  = blockIdx.x * blockDim.x + threadIdx.x;
    int row  = gid >> 5;          // capsule index i
    int lane = gid & 31;          // j
    if (row >= I_SZ) return;
    float v = bias[row * J_SZ + lane];
    float m = v;
    #pragma unroll
    for (int off = 16; off > 0; off >>= 1) m = fmaxf(m, __shfl_xor(m, off, 32));
    float e = __expf(v - m);
    float s = e;
    #pragma unroll
    for (int off = 16; off > 0; off >>= 1) s += __shfl_xor(s, off, 32);
    c[row * J_SZ + lane] = e / s;
}

// ---------------------------------------------------------------------------
// Async staging of one CHUNK: every thread issues exactly 6 b128 async copies
//   A: inputs[b, ibase..+3, 0..15]  ->  lA rows (stride ASTR)
//   B: W[ibase..+3, j, :, :]        ->  lB rows (stride BSTR)
// All global srcs and LDS dsts are 16B aligned by construction.
// ---------------------------------------------------------------------------
__device__ __forceinline__ void stage_chunk_async(const float* __restrict__ inputs,
                                                  const float* __restrict__ W,
                                                  float* lAbuf, float* lBbuf,
                                                  int ibase, int j, int tid) {
#if HAS_ASYNC
    {   // A: thread -> (b = tid>>2, isub = tid&3), 16 floats = 4 x b128
        const int b = tid >> 2, isub = tid & 3;
        const float* src = inputs + ((size_t)b * I_SZ + (ibase + isub)) * P_SZ;
        float* dst = lAbuf + isub * (B_SZ * ASTR) + b * ASTR;
        #pragma unroll
        for (int q = 0; q < 4; ++q)
            g2l_async_b128(src + q * 4, dst + q * 4);
    }
    {   // B: per capsule 128 x 16B chunks, 64 threads/capsule -> 2 x b128 each
        const int isub = tid >> 6, t2 = tid & 63;
        const float* src = W + ((size_t)(ibase + isub) * J_SZ + j) * (P_SZ * D_SZ);
        float* dst = lBbuf + isub * (P_SZ * BSTR);
        #pragma unroll
        for (int q = 0; q < 2; ++q) {
            const int f  = t2 + q * 64;      // 0..127
            const int p  = f >> 3;           // 0..15
            const int dd = (f & 7) * 4;      // 0,4,..,28
            g2l_async_b128(src + p * D_SZ + dd, dst + p * BSTR + dd);
        }
    }
#endif
}

// ---------------------------------------------------------------------------
// Kernel 2: per (j, i-block) partial GEMM  M=64, N=32, K=IB_PER_WG*16
// 8 waves: wave w -> M-tile (w&3), N-tile (w>>2). WMMA f32 16x16x4.
// Async global->LDS double buffering; c[i,j] folded into B fragment.
// ---------------------------------------------------------------------------
__global__ void __launch_bounds__(256)
caps_gemm_kernel(const float* __restrict__ inputs,
                 const float* __restrict__ W,
                 const float* __restrict__ c,
                 float* __restrict__ partial) {
    __shared__ float lA[2][CHUNK * B_SZ * ASTR];   // 2 x 20480 B
    __shared__ float lB[2][CHUNK * P_SZ * BSTR];   // 2 x  9216 B

    const int tid  = threadIdx.x;
    const int lane = tid & 31;
    const int wv   = tid >> 5;        // 0..7
    const int mt   = wv & 3;          // b-tile base = mt*16
    const int nt   = wv >> 2;         // d-tile base = nt*16
    const int j    = blockIdx.x;      // 0..31
    const int i0   = blockIdx.y * IB_PER_WG;
    const int hi   = lane >> 4;       // 0 | 1 (half-wave)
    const int lm   = lane & 15;

    v8f acc = {};

#if HAS_ASYNC
    stage_chunk_async(inputs, W, lA[0], lB[0], i0, j, tid);   // prologue: chunk 0
#endif

    for (int ic = 0; ic < IB_PER_WG; ic += CHUNK) {
        const int ibase = i0 + ic;
        const int cur   = (ic / CHUNK) & 1;

#if HAS_ASYNC
        if (ic + CHUNK < IB_PER_WG) {
            stage_chunk_async(inputs, W, lA[cur ^ 1], lB[cur ^ 1],
                              ibase + CHUNK, j, tid);
            // 6 newest (next chunk) may remain in flight; older chunk complete
            asm volatile("s_wait_asynccnt 6" ::: "memory");
        } else {
            asm volatile("s_wait_asynccnt 0" ::: "memory");
        }
#else
        {   // ---- sync A stage: thread -> (b,isub), 16 floats, float4 LDS stores
            const int b = tid >> 2, isub = tid & 3;
            const float4* src =
                (const float4*)(inputs + ((size_t)b * I_SZ + (ibase + isub)) * P_SZ);
            float4* dst = (float4*)(lA[cur] + isub * (B_SZ * ASTR) + b * ASTR);
            #pragma unroll
            for (int q = 0; q < 4; ++q) dst[q] = src[q];
        }
        {   // ---- sync B stage (raw W; scale applied at fragment read)
            const int isub = tid >> 6, t2 = tid & 63;
            const int p = t2 >> 2, d0 = (t2 & 3) * 8;
            const float* wsrc =
                W + (((size_t)(ibase + isub) * J_SZ + j) * P_SZ + p) * D_SZ + d0;
            if (ic + CHUNK < IB_PER_WG)
                __builtin_prefetch(wsrc + (size_t)CHUNK * J_SZ * P_SZ * D_SZ, 0, 0);
            float4* dst = (float4*)(lB[cur] + isub * (P_SZ * BSTR) + p * BSTR + d0);
            dst[0] = ((const float4*)wsrc)[0];
            dst[1] = ((const float4*)wsrc)[1];
        }
#endif
        // per-chunk routing coefficients (uniform per workgroup)
        float cch[CHUNK];
        #pragma unroll
        for (int u = 0; u < CHUNK; ++u) cch[u] = c[(ibase + u) * J_SZ + j];

        __syncthreads();

        // ---- compute: 4 capsules x 4 K-steps of WMMA f32 16x16x4
        const float* Ab = lA[cur] + (mt * 16 + lm) * ASTR;
        const float* Bb = lB[cur] + nt * 16 + lm;
        #pragma unroll
        for (int isub = 0; isub < CHUNK; ++isub) {
            const float  cij = cch[isub];
            const float* Ai  = Ab + isub * (B_SZ * ASTR);
            const float* Bi  = Bb + isub * (P_SZ * BSTR);
            #pragma unroll
            for (int ks = 0; ks < 4; ++ks) {
                const int pb = ks * 4 + hi * 2;          // K pair {pb, pb+1}
                float2 af = *(const float2*)(Ai + pb);   // 8B-aligned
                v2f Av = {af.x, af.y};
                v2f Bv = {cij * Bi[pb * BSTR], cij * Bi[(pb + 1) * BSTR]};
                acc = __builtin_amdgcn_wmma_f32_16x16x4_f32(
                        false, Av, false, Bv, (short)0, acc, false, false);
            }
        }
        __syncthreads();
    }

    // ---- write partial tile: C/D layout VGPR r -> M = mt*16 + hi*8 + r, N = lm
    const int d = nt * 16 + lm;
    #pragma unroll
    for (int r = 0; r < 8; ++r) {
        const int brow = mt * 16 + hi * 8 + r;
        partial[(size_t)blockIdx.y * (B_SZ * J_SZ * D_SZ)
                + ((size_t)brow * J_SZ + j) * D_SZ + d] = acc[r];
    }
}

// ---------------------------------------------------------------------------
// Kernel 3: reduce NIBLK partials + squash along D.  One wave per (b,j) row.
// ---------------------------------------------------------------------------
__global__ void caps_squash_kernel(const float* __restrict__ partial,
                                   float* __restrict__ out) {
    int gid  = blockIdx.x * blockDim.x + threadIdx.x;
    int row  = gid >> 5;          // b*J + j
    int lane = gid & 31;          // d
    if (row >= B_SZ * J_SZ) return;
    float s = 0.0f;
    #pragma unroll
    for (int k = 0; k < NIBLK; ++k)
        s += partial[(size_t)k * (B_SZ * J_SZ * D_SZ) + (size_t)row * D_SZ + lane];
    float s2 = s * s;
    #pragma unroll
    for (int off = 16; off > 0; off >>= 1) s2 += __shfl_xor(s2, off, 32);
    float scale = s2 / (1.0f + s2) * rsqrtf(s2);    // |v|^2/(1+|v|^2)/|v|
    out[(size_t)row * D_SZ + lane] = scale * s;
}

// ---------------------------------------------------------------------------
extern "C" void kernel_launch(void* const* d_in, const int* in_sizes, int n_in,
                              void* d_out, int out_size, void* d_ws, size_t ws_size,
                              hipStream_t stream) {
    (void)in_sizes; (void)n_in; (void)out_size; (void)ws_size;
    const float* inputs = (const float*)d_in[0];   // [64,2048,16]
    const float* W      = (const float*)d_in[1];   // [2048,32,16,32]
    const float* bias   = (const float*)d_in[2];   // [1,2048,32,1,1]
    float* out = (float*)d_out;                    // [64,32,32]

    float* c       = (float*)d_ws;                 // 2048*32 floats
    float* partial = c + (size_t)I_SZ * J_SZ;      // 16*64*32*32 floats (4 MB)

    caps_softmax_kernel<<<dim3((I_SZ * 32) / 256), dim3(256), 0, stream>>>(bias, c);
    caps_gemm_kernel<<<dim3(J_SZ, NIBLK), dim3(256), 0, stream>>>(inputs, W, c, partial);
    caps_squash_kernel<<<dim3((B_SZ * J_SZ * 32) / 256), dim3(256), 0, stream>>>(partial, out);
}